// PaperInformedMambaBlock_6562710028720
// MI455X (gfx1250) — compile-verified
//
#include <hip/hip_runtime.h>
#include <math.h>

#define B_ 4
#define L_ 4096
#define D_ 1024
#define N_ 32
#define M_ (B_ * L_)          // 16384 rows

#define KT 32                 // K-tile staged in LDS
#define LDST 34               // padded LDS row stride (floats): 32 + 2 (TDM pad)

typedef float v2f __attribute__((ext_vector_type(2)));
typedef float v8f __attribute__((ext_vector_type(8)));
typedef unsigned int v4u __attribute__((ext_vector_type(4)));
typedef int v8i_ __attribute__((ext_vector_type(8)));
typedef int v4i_ __attribute__((ext_vector_type(4)));

#if defined(__has_builtin)
#if __has_builtin(__builtin_amdgcn_wmma_f32_16x16x4_f32)
#define HAVE_WMMA_F32 1
#endif
#if defined(__HIP_DEVICE_COMPILE__) && \
    __has_builtin(__builtin_amdgcn_tensor_load_to_lds) && \
    __has_builtin(__builtin_amdgcn_s_wait_tensorcnt)
#define HAVE_TDM 1
#endif
#endif
#ifndef HAVE_WMMA_F32
#define HAVE_WMMA_F32 0
#endif
#ifndef HAVE_TDM
#define HAVE_TDM 0
#endif

#if HAVE_TDM
// Issue a TDM 2-D tile load: tile_rows x tile_cols fp32 elements, row stride
// row_stride (elements) in global memory, into LDS at lds_addr with 2-DWORD
// padding after every 32 DWORDs (row stride in LDS becomes LDST=34 floats).
// D# packing per CDNA5 ISA.
__device__ __forceinline__
void tdm_load_2d(unsigned lds_addr, const void* gaddr,
                 int tile_cols, int tile_rows, int row_stride)
{
    unsigned long long ga = (unsigned long long)(size_t)gaddr;
    v4u g0;
    g0.x = 1u;                                        // count=1, user mode
    g0.y = lds_addr;                                  // lds_addr (bytes)
    g0.z = (unsigned)(ga & 0xFFFFFFFFu);              // global_addr[31:0]
    g0.w = (unsigned)((ga >> 32) & 0x01FFFFFFu)       // global_addr[56:32]
         | (2u << 30);                                // type=2 ("image")
    v8i_ g1;
    g1[0] = (int)((2u << 16)        // data_size = 4 bytes
                | (1u << 20)        // pad_enable
                | (4u << 22)        // pad_interval: 32 DWORDs
                | (1u << 25));      // pad_amount:   2 DWORDs
    g1[1] = (int)(((unsigned)tile_cols & 0xFFFFu) << 16);          // tensor_dim0[15:0]
    g1[2] = (int)((((unsigned)tile_cols >> 16) & 0xFFFFu)
                | (((unsigned)tile_rows & 0xFFFFu) << 16));        // dim0 hi | dim1 lo
    g1[3] = (int)((((unsigned)tile_rows >> 16) & 0xFFFFu)
                | (((unsigned)tile_cols & 0xFFFFu) << 16));        // dim1 hi | tile_dim0
    g1[4] = (int)((unsigned)tile_rows & 0xFFFFu);                  // tile_dim1 (tile_dim2=0)
    g1[5] = row_stride;                                            // tensor_dim0_stride[31:0]
    g1[6] = 0;                                                     // stride hi | dim1_stride lo
    g1[7] = 0;
    v4i_ z4 = {0, 0, 0, 0};
#if defined(__clang_major__) && (__clang_major__ >= 23)
    v8i_ z8 = {0, 0, 0, 0, 0, 0, 0, 0};
    __builtin_amdgcn_tensor_load_to_lds(g0, g1, z4, z4, z8, 0);
#else
    __builtin_amdgcn_tensor_load_to_lds(g0, g1, z4, z4, 0);
#endif
}
#endif

// ---------------------------------------------------------------------------
// Row-major GEMM: C[m,n] = sum_k A[m,k] * W[n,k] + bias[n] (+resid)
// Block = 256 thr = 8 waves. Block tile 64(M) x 128(N), K staged in KT=32 tiles
// through double-buffered LDS (TDM fills with padded rows -> conflict-free
// ds_load_b64 fragment reads). Wave tile 16x64 (4 f32-WMMA accumulators).
// ---------------------------------------------------------------------------
__global__ __launch_bounds__(256)
void gemm_rtn_kernel(const float* __restrict__ A, const float* __restrict__ W,
                     const float* __restrict__ bias, const float* __restrict__ resid,
                     float* __restrict__ C, int NN, int K)
{
    __shared__ float lds_a[2][64  * LDST];
    __shared__ float lds_w[2][128 * LDST];

    const int lane = threadIdx.x & 31;
    const int wid  = threadIdx.x >> 5;
    const int wm   = wid & 3;                  // 0..3 -> M sub-tile
    const int wn   = wid >> 2;                 // 0..1 -> N sub-group (64 wide)
    const int m0b  = blockIdx.y * 64;
    const int n0b  = blockIdx.x * 128;
    const int half = lane >> 4;                // f32 WMMA K-half
    const int l15  = lane & 15;
    const int NT   = K / KT;

    v8f acc[4] = {};

#if HAVE_TDM
    if (wid == 0) {
        tdm_load_2d((unsigned)(size_t)&lds_a[0][0], A + (size_t)m0b * K, KT, 64,  K);
        tdm_load_2d((unsigned)(size_t)&lds_w[0][0], W + (size_t)n0b * K, KT, 128, K);
    }
#endif

    for (int kt = 0; kt < NT; ++kt) {
        const int buf = kt & 1;
#if HAVE_TDM
        if (wid == 0) {
            if (kt + 1 < NT) {
                const int nb = buf ^ 1;
                tdm_load_2d((unsigned)(size_t)&lds_a[nb][0],
                            A + (size_t)m0b * K + (kt + 1) * KT, KT, 64,  K);
                tdm_load_2d((unsigned)(size_t)&lds_w[nb][0],
                            W + (size_t)n0b * K + (kt + 1) * KT, KT, 128, K);
                __builtin_amdgcn_s_wait_tensorcnt((short)2);   // tile kt complete
            } else {
                __builtin_amdgcn_s_wait_tensorcnt((short)0);
            }
        }
        __syncthreads();
#else
        // Cooperative coalesced fill (fallback / host pass), same padded layout.
        for (int idx = threadIdx.x; idx < 64 * (KT / 4); idx += 256) {
            int r = idx / (KT / 4), c4 = (idx % (KT / 4)) * 4;
            float4 vv = *(const float4*)(A + (size_t)(m0b + r) * K + kt * KT + c4);
            *(float4*)&lds_a[buf][r * LDST + c4] = vv;
        }
        for (int idx = threadIdx.x; idx < 128 * (KT / 4); idx += 256) {
            int r = idx / (KT / 4), c4 = (idx % (KT / 4)) * 4;
            float4 vv = *(const float4*)(W + (size_t)(n0b + r) * K + kt * KT + c4);
            *(float4*)&lds_w[buf][r * LDST + c4] = vv;
        }
        __syncthreads();
#endif

#if HAVE_WMMA_F32
        {
            const float* la = &lds_a[buf][(wm * 16 + l15) * LDST + half * 2];
            const float* lw = &lds_w[buf][(wn * 64 + l15) * LDST + half * 2];
#pragma unroll
            for (int kk = 0; kk < KT; kk += 4) {
                v2f a  = *(const v2f*)(la + kk);
                v2f b0 = *(const v2f*)(lw + kk);
                v2f b1 = *(const v2f*)(lw + 16 * LDST + kk);
                v2f b2 = *(const v2f*)(lw + 32 * LDST + kk);
                v2f b3 = *(const v2f*)(lw + 48 * LDST + kk);
                acc[0] = __builtin_amdgcn_wmma_f32_16x16x4_f32(false, a, false, b0, (short)0, acc[0], false, false);
                acc[1] = __builtin_amdgcn_wmma_f32_16x16x4_f32(false, a, false, b1, (short)0, acc[1], false, false);
                acc[2] = __builtin_amdgcn_wmma_f32_16x16x4_f32(false, a, false, b2, (short)0, acc[2], false, false);
                acc[3] = __builtin_amdgcn_wmma_f32_16x16x4_f32(false, a, false, b3, (short)0, acc[3], false, false);
            }
        }
#else
        for (int sub = 0; sub < 4; ++sub)
            for (int r = 0; r < 8; ++r) {
                int rl = wm * 16 + r + half * 8;
                int cl = wn * 64 + sub * 16 + l15;
                float sv = acc[sub][r];
                for (int kk = 0; kk < KT; ++kk)
                    sv += lds_a[buf][rl * LDST + kk] * lds_w[buf][cl * LDST + kk];
                acc[sub][r] = sv;
            }
#endif
        __syncthreads();   // protect buf before it is refilled (2 tiles later)
    }

    for (int sub = 0; sub < 4; ++sub) {
        int col = n0b + wn * 64 + sub * 16 + l15;
        float bv = bias[col];
        for (int r = 0; r < 8; ++r) {
            int row = m0b + wm * 16 + r + half * 8;
            float v = acc[sub][r] + bv;
            if (resid) v += resid[(size_t)row * NN + col];
            C[(size_t)row * NN + col] = v;
        }
    }
}

// ---------------------------------------------------------------------------
// Depthwise conv(k=4, pad 2) + SiLU.  Reads xa half of xz, writes xc (t<L).
// ---------------------------------------------------------------------------
__global__ __launch_bounds__(256)
void conv_silu_kernel(const float* __restrict__ xz, const float* __restrict__ cw,
                      const float* __restrict__ cb, float* __restrict__ xc)
{
    size_t i = (size_t)blockIdx.x * 256 + threadIdx.x;   // over B*L*D
    if (i >= (size_t)M_ * D_) return;
    int d = (int)(i % D_);
    size_t bt = i / D_;
    int t = (int)(bt % L_);
    int b = (int)(bt / L_);
    float s = cb[d];
#pragma unroll
    for (int k = 0; k < 4; ++k) {
        int tt = t - 2 + k;
        if (tt >= 0 && tt < L_)
            s += xz[((size_t)b * L_ + tt) * (2 * D_) + d] * cw[d * 4 + k];
    }
    float sig = 1.f / (1.f + __expf(-s));
    xc[i] = s * sig;
}

// ---------------------------------------------------------------------------
// B/C projections: Bm,Cm [M,32] = xc[M,1024] @ {B_w,C_w}^T + bias.  WMMA,
// register software pipeline (load k+4 fragments while k is in the matrix pipe).
// ---------------------------------------------------------------------------
__global__ __launch_bounds__(256)
void gemm_bc_kernel(const float* __restrict__ xc,
                    const float* __restrict__ Bw, const float* __restrict__ Bb,
                    const float* __restrict__ Cw, const float* __restrict__ Cb,
                    float* __restrict__ Bm, float* __restrict__ Cm)
{
    const int lane = threadIdx.x & 31;
    const int wid  = threadIdx.x >> 5;
    const int half = lane >> 4;
    const int l15  = lane & 15;
    const int tile = blockIdx.x;                    // 0..3
    const int m0   = (blockIdx.y * 8 + wid) * 16;

    const float* Wsel = (tile < 2) ? Bw : Cw;
    const float* bsel = (tile < 2) ? Bb : Cb;
    float*       Osel = (tile < 2) ? Bm : Cm;
    const int ncol0   = (tile & 1) * 16;

    v8f acc = {};
#if HAVE_WMMA_F32
    const float* arow = xc   + (size_t)(m0 + l15) * D_ + half * 2;
    const float* wrow = Wsel + (size_t)(ncol0 + l15) * D_ + half * 2;
    v2f a = *(const v2f*)(arow);
    v2f b = *(const v2f*)(wrow);
    for (int k = 4; k <= D_; k += 4) {
        v2f an = a, bn = b;
        if (k < D_) { an = *(const v2f*)(arow + k); bn = *(const v2f*)(wrow + k); }
        acc = __builtin_amdgcn_wmma_f32_16x16x4_f32(false, a, false, b, (short)0, acc, false, false);
        a = an; b = bn;
    }
#else
    for (int r = 0; r < 8; ++r) {
        int row = m0 + r + half * 8;
        int col = ncol0 + l15;
        float sv = 0.f;
        for (int k = 0; k < D_; ++k) sv += xc[(size_t)row * D_ + k] * Wsel[(size_t)col * D_ + k];
        acc[r] = sv;
    }
#endif
    float bv = bsel[ncol0 + l15];
    for (int r = 0; r < 8; ++r) {
        int row = m0 + r + half * 8;
        Osel[(size_t)row * N_ + ncol0 + l15] = acc[r] + bv;
    }
}

// ---------------------------------------------------------------------------
// Selective scan. lane = n (N=32 == wave32). Each wave owns 4 channels d.
// s = clip(s*A + b_t*x_t, -10, 10); y = sum_n c_t*s;  out = y * silu(z),
// written in-place over xc (read-before-write, single-owner per (b,t,d)).
// ---------------------------------------------------------------------------
__global__ __launch_bounds__(256)
void scan_kernel(const float* __restrict__ xz, float* xc /* in: xc, out: y2 */,
                 const float* __restrict__ Bm, const float* __restrict__ Cm,
                 const float* __restrict__ A_log)
{
    const int lane = threadIdx.x & 31;           // n
    const int wid  = threadIdx.x >> 5;
    const int b    = blockIdx.x / (D_ / 32);
    const int dg   = blockIdx.x % (D_ / 32);
    const int d0   = dg * 32 + wid * 4;

    float Av[4], s[4];
#pragma unroll
    for (int i = 0; i < 4; ++i) {
        float al = A_log[(d0 + i) * N_ + lane];
        al = fminf(fmaxf(al, -5.f), 2.f);
        Av[i] = -__expf(al);
        s[i] = 0.f;
    }

    for (int t = 0; t < L_; ++t) {
        size_t m = (size_t)b * L_ + t;
        float bb = Bm[m * N_ + lane];
        float cc = Cm[m * N_ + lane];
        float4 xv = *(const float4*)(xc + m * D_ + d0);            // broadcast
        float4 zv = *(const float4*)(xz + m * (2 * D_) + D_ + d0); // z half
        float xs[4] = { xv.x, xv.y, xv.z, xv.w };
        float zs[4] = { zv.x, zv.y, zv.z, zv.w };
        float y[4];
#pragma unroll
        for (int i = 0; i < 4; ++i) {
            float sv = s[i] * Av[i] + bb * xs[i];
            sv = fminf(fmaxf(sv, -10.f), 10.f);
            s[i] = sv;
            y[i] = cc * sv;
        }
#pragma unroll
        for (int i = 0; i < 4; ++i) {
            float v = y[i];
#pragma unroll
            for (int off = 16; off >= 1; off >>= 1) v += __shfl_xor(v, off, 32);
            y[i] = v;
        }
        if (lane == 0) {
            float4 o;
            float* op = &o.x;
#pragma unroll
            for (int i = 0; i < 4; ++i) {
                float sig = 1.f / (1.f + __expf(-zs[i]));
                op[i] = y[i] * zs[i] * sig;                        // y * silu(z)
            }
            *(float4*)(xc + m * D_ + d0) = o;
        }
    }
}

// ---------------------------------------------------------------------------
// LayerNorm over D=1024, one block (256 thr, float4 each) per row, in-place.
// ---------------------------------------------------------------------------
__global__ __launch_bounds__(256)
void ln_kernel(float* __restrict__ h, const float* __restrict__ lnw,
               const float* __restrict__ lnb)
{
    __shared__ float red[2][8];
    const int lane = threadIdx.x & 31;
    const int wid  = threadIdx.x >> 5;
    float* p = h + (size_t)blockIdx.x * D_;
    const int c = threadIdx.x * 4;

    float4 v = *(const float4*)(p + c);
    float lsum = v.x + v.y + v.z + v.w;
    float lsq  = v.x * v.x + v.y * v.y + v.z * v.z + v.w * v.w;
#pragma unroll
    for (int off = 16; off >= 1; off >>= 1) {
        lsum += __shfl_xor(lsum, off, 32);
        lsq  += __shfl_xor(lsq,  off, 32);
    }
    if (lane == 0) { red[0][wid] = lsum; red[1][wid] = lsq; }
    __syncthreads();
    float tsum = 0.f, tsq = 0.f;
#pragma unroll
    for (int w = 0; w < 8; ++w) { tsum += red[0][w]; tsq += red[1][w]; }
    float mu   = tsum * (1.f / D_);
    float var  = tsq * (1.f / D_) - mu * mu;
    float rstd = rsqrtf(var + 1e-5f);

    float4 w4 = *(const float4*)(lnw + c);
    float4 b4 = *(const float4*)(lnb + c);
    float4 o;
    o.x = (v.x - mu) * rstd * w4.x + b4.x;
    o.y = (v.y - mu) * rstd * w4.y + b4.y;
    o.z = (v.z - mu) * rstd * w4.z + b4.z;
    o.w = (v.w - mu) * rstd * w4.w + b4.w;
    *(float4*)(p + c) = o;
}

// ---------------------------------------------------------------------------
extern "C" void kernel_launch(void* const* d_in, const int* in_sizes, int n_in,
                              void* d_out, int out_size, void* d_ws, size_t ws_size,
                              hipStream_t stream)
{
    const float* x      = (const float*)d_in[0];
    const float* in_w   = (const float*)d_in[1];
    const float* in_b   = (const float*)d_in[2];
    const float* conv_w = (const float*)d_in[3];
    const float* conv_b = (const float*)d_in[4];
    const float* A_log  = (const float*)d_in[5];
    const float* B_w    = (const float*)d_in[6];
    const float* B_b    = (const float*)d_in[7];
    const float* C_w    = (const float*)d_in[8];
    const float* C_b    = (const float*)d_in[9];
    const float* out_w  = (const float*)d_in[10];
    const float* out_b  = (const float*)d_in[11];
    const float* ln_w   = (const float*)d_in[12];
    const float* ln_b   = (const float*)d_in[13];
    float* out = (float*)d_out;

    float* ws = (float*)d_ws;
    float* xz = ws;                                 // M * 2D
    float* xc = xz + (size_t)M_ * 2 * D_;           // M * D (later reused as y2)
    float* Bm = xc + (size_t)M_ * D_;               // M * N
    float* Cm = Bm + (size_t)M_ * N_;               // M * N

    // 1) xz = x @ in_w^T + in_b
    dim3 g1(2 * D_ / 128, M_ / 64);                 // (16, 256)
    gemm_rtn_kernel<<<g1, 256, 0, stream>>>(x, in_w, in_b, nullptr, xz, 2 * D_, D_);

    // 2) xc = silu(depthwise_conv(xa) + conv_b)
    size_t tot = (size_t)M_ * D_;
    conv_silu_kernel<<<(unsigned)((tot + 255) / 256), 256, 0, stream>>>(xz, conv_w, conv_b, xc);

    // 3) Bm, Cm projections
    dim3 g3(4, M_ / 128);
    gemm_bc_kernel<<<g3, 256, 0, stream>>>(xc, B_w, B_b, C_w, C_b, Bm, Cm);

    // 4) selective scan + y*silu(z), in-place xc -> y2
    scan_kernel<<<B_ * D_ / 32, 256, 0, stream>>>(xz, xc, Bm, Cm, A_log);

    // 5) h = y2 @ out_w^T + out_b + x  -> d_out
    dim3 g5(D_ / 128, M_ / 64);                     // (8, 256)
    gemm_rtn_kernel<<<g5, 256, 0, stream>>>(xc, out_w, out_b, x, out, D_, D_);

    // 6) LayerNorm in-place on d_out
    ln_kernel<<<M_, 256, 0, stream>>>(out, ln_w, ln_b);
}